// CSD_3272765080047
// MI455X (gfx1250) — compile-verified
//
#include <hip/hip_runtime.h>

#define NUM_CLASSES 10
#define THREADS 256
#define TILE_ROWS 1024
#define TILE_FLOATS (TILE_ROWS * NUM_CLASSES)          // 10240 floats = 40 KB staged per block
#define CHUNKS_PER_THREAD (TILE_FLOATS / 4 / THREADS)  // 10 x b128 async loads per thread
#define SUBTILES_PER_WAVE (TILE_ROWS / 16 / 8)         // 8 waves, 16-row WMMA subtiles -> 8

typedef float v2f __attribute__((ext_vector_type(2)));
typedef float v8f __attribute__((ext_vector_type(8)));

// Workspace layout (d_ws):
//   bytes [0,40)   : unsigned counts[10]
//   bytes [64,72)  : double ssq   (sum of squares of outputs)
//   bytes [72,80)  : double slog2 (sum of log2(nominator_i))

__global__ void csd_init_ws(unsigned* __restrict__ counts, double* __restrict__ acc) {
  if (threadIdx.x < NUM_CLASSES) counts[threadIdx.x] = 0u;
  if (threadIdx.x == 0) { acc[0] = 0.0; acc[1] = 0.0; }
}

__global__ void csd_count(const int* __restrict__ target,
                          unsigned* __restrict__ counts, int n) {
  __shared__ unsigned hist[NUM_CLASSES];
  if (threadIdx.x < NUM_CLASSES) hist[threadIdx.x] = 0u;
  __syncthreads();

  const int n4 = n >> 2;
  const int4* t4 = (const int4*)target;
  const int stride = gridDim.x * blockDim.x;
  for (int i = blockIdx.x * blockDim.x + threadIdx.x; i < n4; i += stride) {
    int4 v = t4[i];
    atomicAdd(&hist[v.x], 1u);
    atomicAdd(&hist[v.y], 1u);
    atomicAdd(&hist[v.z], 1u);
    atomicAdd(&hist[v.w], 1u);
  }
  if (blockIdx.x == 0 && threadIdx.x == 0) {
    for (int i = n4 << 2; i < n; ++i) atomicAdd(&counts[target[i]], 1u);
  }
  __syncthreads();
  if (threadIdx.x < NUM_CLASSES) atomicAdd(&counts[threadIdx.x], hist[threadIdx.x]);
}

__global__ void __launch_bounds__(THREADS)
csd_main(const float* __restrict__ outputs,
         const unsigned* __restrict__ counts,
         double* __restrict__ acc, int nrows) {
  __shared__ float smem[TILE_FLOATS + 1];   // +1 : zero pad slot for K>=10
  __shared__ float red_ssq[THREADS];
  __shared__ float red_slog[THREADS];

  float cf[NUM_CLASSES];
#pragma unroll
  for (int k = 0; k < NUM_CLASSES; ++k) cf[k] = (float)counts[k];

  const unsigned tid = threadIdx.x;
  const int rowBase = blockIdx.x * TILE_ROWS;

  float ssq = 0.0f;
  float slog2 = 0.0f;   // accumulated in log2 domain; x ln2 once in finalize

  if (rowBase + TILE_ROWS <= nrows) {
    // ---- fast path: async-stage 40 KB of rows into LDS (CDNA5 async path) ----
    unsigned long long gaddr =
        (unsigned long long)(uintptr_t)(outputs + (size_t)rowBase * NUM_CLASSES) +
        (unsigned long long)tid * 16ull;
    unsigned laddr = (unsigned)(uintptr_t)(void*)smem + tid * 16u;
#pragma unroll
    for (int j = 0; j < CHUNKS_PER_THREAD; ++j) {
      asm volatile("global_load_async_to_lds_b128 %0, %1, off"
                   :: "v"(laddr), "v"(gaddr)
                   : "memory");
      gaddr += (unsigned long long)THREADS * 16ull;
      laddr += THREADS * 16u;
    }
    if (tid == 0) smem[TILE_FLOATS] = 0.0f;   // zero pad slot (DScnt, not ASYNCcnt)
    asm volatile("s_wait_asynccnt 0" ::: "memory");
    __syncthreads();

    // ---- WMMA GEMV: nominator(16 rows) = A(16x10) x counts, via 3x 16x16x4 f32 ----
    const unsigned lane = tid & 31u;
    const unsigned wave = tid >> 5;
    const int m16 = (int)(lane & 15u);
    const bool hi = (lane & 16u) != 0u;

    // B fragments: B[k][n] = counts[k] for every column n
    v2f b0, b1, b2;
    b0.x = hi ? cf[2] : cf[0];  b0.y = hi ? cf[3] : cf[1];
    b1.x = hi ? cf[6] : cf[4];  b1.y = hi ? cf[7] : cf[5];
    b2.x = hi ? 0.0f  : cf[8];  b2.y = hi ? 0.0f  : cf[9];

    for (int s = 0; s < SUBTILES_PER_WAVE; ++s) {
      const int row0 = ((int)wave * SUBTILES_PER_WAVE + s) * 16;
      const int rbase = (row0 + m16) * NUM_CLASSES;

      v2f a0, a1, a2;
      a0.x = smem[rbase + (hi ? 2 : 0)];
      a0.y = smem[rbase + (hi ? 3 : 1)];
      a1.x = smem[rbase + (hi ? 6 : 4)];
      a1.y = smem[rbase + (hi ? 7 : 5)];
      const int i8 = hi ? TILE_FLOATS : rbase + 8;   // zero slot for K=10,11
      const int i9 = hi ? TILE_FLOATS : rbase + 9;
      a2.x = smem[i8];
      a2.y = smem[i9];

      // ssq: the wave's A fragments collectively touch each element exactly once
      ssq = fmaf(a0.x, a0.x, ssq);
      ssq = fmaf(a0.y, a0.y, ssq);
      ssq = fmaf(a1.x, a1.x, ssq);
      ssq = fmaf(a1.y, a1.y, ssq);
      ssq = fmaf(a2.x, a2.x, ssq);
      ssq = fmaf(a2.y, a2.y, ssq);

      v8f accv = {};
      accv = __builtin_amdgcn_wmma_f32_16x16x4_f32(false, a0, false, b0,
                                                   (short)0, accv, false, false);
      accv = __builtin_amdgcn_wmma_f32_16x16x4_f32(false, a1, false, b1,
                                                   (short)0, accv, false, false);
      accv = __builtin_amdgcn_wmma_f32_16x16x4_f32(false, a2, false, b2,
                                                   (short)0, accv, false, false);

      // D layout: VGPR r, lane L -> M = r + (L<16?0:8), N = L%16.
      // Lanes with N==0 (lane 0 and 16) hold all 16 nominators between them.
      // nominator ~ [1e4, 2e6]: never denormal/inf -> raw v_log_f32 is exact enough.
      if (m16 == 0) {
#pragma unroll
        for (int r = 0; r < 8; ++r) slog2 += __builtin_amdgcn_logf(accv[r]);
      }
    }
  } else {
    // ---- tail tile: guarded direct global reads, scalar path (no WMMA: EXEC varies) ----
    for (int row = rowBase + (int)tid; row < nrows; row += THREADS) {
      const float* rp = outputs + (size_t)row * NUM_CLASSES;
      float nom = 0.0f;
#pragma unroll
      for (int k = 0; k < NUM_CLASSES; ++k) {
        float x = rp[k];
        nom = fmaf(x, cf[k], nom);
        ssq = fmaf(x, x, ssq);
      }
      slog2 += __builtin_amdgcn_logf(nom);
    }
    __syncthreads();  // keep barrier count consistent with fast path
  }

  // ---- block reduction ----
  red_ssq[tid]  = ssq;
  red_slog[tid] = slog2;
  __syncthreads();
#pragma unroll
  for (int s = THREADS / 2; s > 0; s >>= 1) {
    if (tid < (unsigned)s) {
      red_ssq[tid]  += red_ssq[tid + s];
      red_slog[tid] += red_slog[tid + s];
    }
    __syncthreads();
  }
  if (tid == 0) {
    atomicAdd(&acc[0], (double)red_ssq[0]);
    atomicAdd(&acc[1], (double)red_slog[0]);
  }
}

__global__ void csd_finalize(const double* __restrict__ acc,
                             float* __restrict__ out, int nrows) {
  double ssq   = acc[0];
  double slog2 = acc[1];
  double n = (double)nrows;
  // mean(-log(nom/denom)) = log(denom) - mean(ln nom),
  // log(denom) = 0.5*log(ssq) + 0.5*log(N);  ln nom = ln2 * log2 nom
  const double LN2 = 0.6931471805599453;
  double res = 0.5 * log(ssq) + 0.5 * log(n) - (slog2 * LN2) / n;
  out[0] = (float)res;
}

extern "C" void kernel_launch(void* const* d_in, const int* in_sizes, int n_in,
                              void* d_out, int out_size, void* d_ws, size_t ws_size,
                              hipStream_t stream) {
  const float* outputs = (const float*)d_in[0];
  const int* target = (const int*)d_in[1];
  const int nrows = in_sizes[0] / NUM_CLASSES;

  unsigned* counts = (unsigned*)d_ws;
  double* acc = (double*)((char*)d_ws + 64);
  float* out = (float*)d_out;

  csd_init_ws<<<1, 32, 0, stream>>>(counts, acc);

  int cblocks = (nrows / 4 + THREADS - 1) / THREADS;
  if (cblocks < 1) cblocks = 1;
  if (cblocks > 2048) cblocks = 2048;
  csd_count<<<cblocks, THREADS, 0, stream>>>(target, counts, nrows);

  int mblocks = (nrows + TILE_ROWS - 1) / TILE_ROWS;
  if (mblocks < 1) mblocks = 1;
  csd_main<<<mblocks, THREADS, 0, stream>>>(outputs, counts, acc, nrows);

  csd_finalize<<<1, 1, 0, stream>>>(acc, out, nrows);
}